// BahdanauAttention_32942399160628
// MI455X (gfx1250) — compile-verified
//
#include <hip/hip_runtime.h>
#include <cmath>

#define B_ 32
#define S_ 2048
#define H_ 1024

typedef __attribute__((ext_vector_type(2))) float v2f;
typedef __attribute__((ext_vector_type(8))) float v8f;

// ---------------------------------------------------------------------------
// Kernel 0: qb[b][o] = dot(query[b,:], Wa_w[o,:]) + Wa_b[o] + Ua_b[o]
// Tiny (67 MFLOP); query row cached in LDS, Wa_w streamed with float4 loads.
// ---------------------------------------------------------------------------
__global__ __launch_bounds__(256) void qb_kernel(
    const float* __restrict__ query, const float* __restrict__ Wa_w,
    const float* __restrict__ Wa_b, const float* __restrict__ Ua_b,
    float* __restrict__ qb) {
  __shared__ float qrow[H_];
  const int b = blockIdx.y;
  const int o = blockIdx.x * 256 + threadIdx.x;
  for (int i = threadIdx.x; i < H_; i += 256) qrow[i] = query[b * H_ + i];
  __syncthreads();
  float acc = Wa_b[o] + Ua_b[o];
  const float* wr = Wa_w + (size_t)o * H_;
#pragma unroll 4
  for (int h = 0; h < H_; h += 4) {
    float4 w4 = *(const float4*)(wr + h);
    acc += qrow[h] * w4.x + qrow[h + 1] * w4.y + qrow[h + 2] * w4.z +
           qrow[h + 3] * w4.w;
  }
  qb[b * H_ + o] = acc;
}

// ---------------------------------------------------------------------------
// Kernel 1 (the big one): fused  scores[b,s] = Va . tanh(keys[b,s,:]Ua^T + qb[b,:])
// Workgroup = 512 threads = 16 waves, owns a 64-row x 1024-col tile.
// Each wave holds a 4x4 register tile (4 row-subtiles x 4 n-tiles,
// 16 f32 WMMA accumulators = 128 VGPRs): per 4-K step only 4 A-frag +
// 4 B-frag b64 loads feed 16 v_wmma_f32_16x16x4_f32 ops (8 loads/16 WMMA).
// All 16 waves share the same 64 keys rows (A-frags hit WGP$); each Ua_w
// fragment is loaded by exactly one wave. h-tile never leaves registers.
// ---------------------------------------------------------------------------
__global__ __launch_bounds__(512) void score_kernel(
    const float* __restrict__ keys, const float* __restrict__ Ua_w,
    const float* __restrict__ qb, const float* __restrict__ Va_w,
    const float* __restrict__ Va_b, float* __restrict__ scores) {
  __shared__ float srow[64];
  const int tid = threadIdx.x;
  const int lane = tid & 31;
  const int wv = tid >> 5;    // 0..15: this wave's 64-column group
  const int ln = lane & 15;
  const int khalf = (lane < 16) ? 0 : 2;  // K half per WMMA A/B fragment layout
  const int stile = blockIdx.x;           // 32 tiles of 64 rows
  const int b = blockIdx.y;

  if (tid < 64) srow[tid] = 0.0f;
  __syncthreads();

  // A fragments: rows (stile*64 + mt*16 + ln) of keys, contiguous float2.
  const float* arow0 =
      keys + ((size_t)(b * S_ + stile * 64 + ln)) * H_ + khalf;
  // B fragments: B[k][n] = Ua_w[n*H + k]; this wave owns n-tiles wv*4..wv*4+3.
  const float* brow0 = Ua_w + ((size_t)(wv * 64 + ln)) * H_ + khalf;

  v8f acc[16];  // acc[mt*4+t] : row-subtile mt, n-tile t
#pragma unroll
  for (int i = 0; i < 16; ++i) acc[i] = (v8f){};

#pragma unroll 1
  for (int kc = 0; kc < H_; kc += 32) {
    // Pull the next chunk of each streamed row toward the WGP
    // (global_prefetch_b8; speculative, OOB-safe).
    if (kc + 160 <= H_) {
#pragma unroll
      for (int mt = 0; mt < 4; ++mt)
        __builtin_prefetch(arow0 + (size_t)mt * 16 * H_ + kc + 128);
#pragma unroll
      for (int t = 0; t < 4; ++t)
        __builtin_prefetch(brow0 + (size_t)t * 16 * H_ + kc + 128);
    }
#pragma unroll
    for (int kk = 0; kk < 32; kk += 4) {
      const int k0 = kc + kk;
      v2f a[4], bv[4];
#pragma unroll
      for (int mt = 0; mt < 4; ++mt)
        a[mt] = *(const v2f*)(arow0 + (size_t)mt * 16 * H_ + k0);
#pragma unroll
      for (int t = 0; t < 4; ++t)
        bv[t] = *(const v2f*)(brow0 + (size_t)t * 16 * H_ + k0);
#pragma unroll
      for (int mt = 0; mt < 4; ++mt)
#pragma unroll
        for (int t = 0; t < 4; ++t)
          acc[mt * 4 + t] = __builtin_amdgcn_wmma_f32_16x16x4_f32(
              false, a[mt], false, bv[t], (short)0, acc[mt * 4 + t], false,
              false);
    }
  }

  // Epilogue: h = tanh(acc + qb[n]); partial score = sum_n h * Va[n].
  // C/D layout: lanes 0-15 hold (M=r, N=ln); lanes 16-31 hold (M=r+8, N=ln).
  float part[4][8];
#pragma unroll
  for (int mt = 0; mt < 4; ++mt)
#pragma unroll
    for (int r = 0; r < 8; ++r) part[mt][r] = 0.0f;
#pragma unroll
  for (int t = 0; t < 4; ++t) {
    const int n = wv * 64 + t * 16 + ln;
    const float qv = qb[b * H_ + n];  // includes Wa_b + Ua_b
    const float vv = Va_w[n];
#pragma unroll
    for (int mt = 0; mt < 4; ++mt)
#pragma unroll
      for (int r = 0; r < 8; ++r)
        part[mt][r] += tanhf(acc[mt * 4 + t][r] + qv) * vv;
  }
  // Reduce over the 16 lanes of each half (N dimension) via xor shuffles.
#pragma unroll
  for (int off = 1; off < 16; off <<= 1) {
#pragma unroll
    for (int mt = 0; mt < 4; ++mt)
#pragma unroll
      for (int r = 0; r < 8; ++r)
        part[mt][r] += __shfl_xor(part[mt][r], off, 32);
  }
  if (ln == 0) {
    const int half = (lane < 16) ? 0 : 8;
#pragma unroll
    for (int mt = 0; mt < 4; ++mt)
#pragma unroll
      for (int r = 0; r < 8; ++r)
        atomicAdd(&srow[mt * 16 + half + r], part[mt][r]);
  }
  __syncthreads();
  if (tid < 64)
    scores[(size_t)b * S_ + stile * 64 + tid] = srow[tid] + Va_b[0];
}

// ---------------------------------------------------------------------------
// Kernel 2: per-batch softmax over S=2048 scores (one block per batch).
// ---------------------------------------------------------------------------
__global__ __launch_bounds__(256) void softmax_kernel(
    const float* __restrict__ scores, float* __restrict__ weights) {
  __shared__ float red[256];
  const int b = blockIdx.x, tid = threadIdx.x;
  const float* sr = scores + (size_t)b * S_;
  float v[8];
  float lmax = -INFINITY;
#pragma unroll
  for (int i = 0; i < 8; ++i) {
    v[i] = sr[tid + 256 * i];
    lmax = fmaxf(lmax, v[i]);
  }
  red[tid] = lmax;
  __syncthreads();
  for (int o = 128; o > 0; o >>= 1) {
    if (tid < o) red[tid] = fmaxf(red[tid], red[tid + o]);
    __syncthreads();
  }
  const float m = red[0];
  __syncthreads();
  float lsum = 0.0f;
#pragma unroll
  for (int i = 0; i < 8; ++i) {
    v[i] = expf(v[i] - m);
    lsum += v[i];
  }
  red[tid] = lsum;
  __syncthreads();
  for (int o = 128; o > 0; o >>= 1) {
    if (tid < o) red[tid] += red[tid + o];
    __syncthreads();
  }
  const float inv = 1.0f / red[0];
#pragma unroll
  for (int i = 0; i < 8; ++i)
    weights[(size_t)b * S_ + tid + 256 * i] = v[i] * inv;
}

// ---------------------------------------------------------------------------
// Kernel 3: context[b][h] = sum_s weights[b][s] * keys[b][s][h].
// Coalesced over h; weight chunks staged in LDS. Pure-bandwidth (256 MB).
// ---------------------------------------------------------------------------
__global__ __launch_bounds__(256) void context_kernel(
    const float* __restrict__ keys, const float* __restrict__ weights,
    float* __restrict__ context) {
  __shared__ float wch[256];
  const int b = blockIdx.y;
  const int h = blockIdx.x * 256 + threadIdx.x;
  const float* kb = keys + (size_t)b * S_ * H_ + h;
  float acc = 0.0f;
  for (int s0 = 0; s0 < S_; s0 += 256) {
    __syncthreads();
    wch[threadIdx.x] = weights[(size_t)b * S_ + s0 + threadIdx.x];
    __syncthreads();
#pragma unroll 4
    for (int j = 0; j < 256; ++j) acc += wch[j] * kb[(size_t)(s0 + j) * H_];
  }
  context[b * H_ + h] = acc;
}

// ---------------------------------------------------------------------------
extern "C" void kernel_launch(void* const* d_in, const int* in_sizes, int n_in,
                              void* d_out, int out_size, void* d_ws,
                              size_t ws_size, hipStream_t stream) {
  const float* query = (const float*)d_in[0];
  const float* keys  = (const float*)d_in[1];
  const float* Wa_w  = (const float*)d_in[2];
  const float* Wa_b  = (const float*)d_in[3];
  const float* Ua_w  = (const float*)d_in[4];
  const float* Ua_b  = (const float*)d_in[5];
  const float* Va_w  = (const float*)d_in[6];
  const float* Va_b  = (const float*)d_in[7];

  float* out     = (float*)d_out;
  float* context = out;            // (B,1,H) = 32768 floats
  float* weights = out + B_ * H_;  // (B,1,S) = 65536 floats

  float* qb     = (float*)d_ws;    // 32768 floats
  float* scores = qb + B_ * H_;    // 65536 floats

  qb_kernel<<<dim3(H_ / 256, B_), 256, 0, stream>>>(query, Wa_w, Wa_b, Ua_b, qb);
  score_kernel<<<dim3(S_ / 64, B_), 512, 0, stream>>>(keys, Ua_w, qb, Va_w,
                                                      Va_b, scores);
  softmax_kernel<<<B_, 256, 0, stream>>>(scores, weights);
  context_kernel<<<dim3(H_ / 256, B_), 256, 0, stream>>>(keys, weights,
                                                         context);
}